// EvolvedNetwork_90117003804689
// MI455X (gfx1250) — compile-verified
//
#include <hip/hip_runtime.h>

typedef __attribute__((ext_vector_type(2))) float v2f;
typedef __attribute__((ext_vector_type(4))) float v4f;
typedef __attribute__((ext_vector_type(8))) float v8f;

#define B_ROWS   131072
#define D0       784
#define N1       50
#define N1P      64
#define N2       30
#define N2P      32
#define N3       10
#define N3P      16
#define NO       10
#define NOP      16

#define MB       64     // rows per workgroup (4 waves x 16)
#define NCH      7      // K chunks for layer 1
#define KC       112    // 784 = 7 * 112, 112 % 4 == 0
#define XP       116    // xs pitch: 116 mod 64 = 52, gcd(52,64)=4 -> 16 distinct even banks
#define W1PP     160    // pair-pitch (2 K-rows interleaved, 128 used): 160 mod 64 = 32
#define H1P      68     // 68 mod 64 = 4 -> conflict-free b64 A loads
#define W2PP     96     // 64 used, 96 mod 64 = 32
#define H2P      36
#define W3PP     96     // 32 used
#define H3P      20
#define WOPP     96     // 32 used

__device__ __forceinline__ v8f wmma4(v2f a, v2f b, v8f c) {
  // D = A(16x4 f32) * B(4x16 f32) + C(16x16 f32), full fp32 precision
  return __builtin_amdgcn_wmma_f32_16x16x4_f32(false, a, false, b, (short)0, c, false, false);
}

__device__ __forceinline__ float act_apply(int a, float z) {
  // a==0: relu ; a==1: tanh(z) = 2*sigmoid(2z)-1 ; a==2: sigmoid(z)
  // one v_exp_f32 + one v_rcp_f32
  float scale = (a == 1) ? 2.0f : 1.0f;
  float e = __expf(-scale * z);
  float r = scale * __builtin_amdgcn_rcpf(1.0f + e) - (scale - 1.0f);
  return (a == 0) ? fmaxf(z, 0.0f) : r;
}

__launch_bounds__(128)
__global__ void mlp4_wmma_kernel(const float* __restrict__ x,
    const float* __restrict__ W1, const float* __restrict__ b1, const int* __restrict__ a1,
    const float* __restrict__ W2, const float* __restrict__ b2, const int* __restrict__ a2,
    const float* __restrict__ W3, const float* __restrict__ b3, const int* __restrict__ a3,
    const float* __restrict__ Wo, const float* __restrict__ bo,
    float* __restrict__ out)
{
  // ~118 KB total -> 2 workgroups per WGP (320 KB LDS)
  __shared__ __attribute__((aligned(16))) float xs [MB * XP];           // 29,696 B
  __shared__ __attribute__((aligned(16))) float w1t[(KC / 2) * W1PP];   // 35,840 B
  __shared__ __attribute__((aligned(16))) float h1s[MB * H1P];          // 17,408 B
  __shared__ __attribute__((aligned(16))) float w2t[(N1P / 2) * W2PP];  // 12,288 B
  __shared__ __attribute__((aligned(16))) float h2s[MB * H2P];          //  9,216 B
  __shared__ __attribute__((aligned(16))) float w3t[(N2P / 2) * W3PP];  //  6,144 B
  __shared__ __attribute__((aligned(16))) float h3s[MB * H3P];          //  5,120 B
  __shared__ __attribute__((aligned(16))) float wot[(N3P / 2) * WOPP];  //  3,072 B
  __shared__ float bb1[N1P], bb2[N2P], bb3[N3P], bbo[NOP];
  __shared__ int   aa1[N1P], aa2[N2P], aa3[N3P];

  const int tid  = threadIdx.x;
  const int wave = tid >> 5;
  const int lane = tid & 31;
  const int n16  = lane & 15;            // N index (B/C/D row-striped layout)
  const int khi  = (lane >> 4) << 1;     // K offset 0/2 for A fragments
  const int phi  = (lane >> 4);          // K-pair offset 0/1 for B fragments
  const int mhi  = (lane >> 4) << 3;     // M offset 0/8 for C/D layout
  const long row0 = (long)blockIdx.x * MB;

  // -------- stage small weights in fragment-ready (K-pair interleaved) layout -----
  for (int i = tid; i < (N1P / 2) * N2P; i += 128) {      // 8 iters
    int p = i >> 5, n = i & 31;
    int k0 = 2 * p;
    v2f w;
    w.x = (k0     < N1 && n < N2) ? W2[k0 * N2 + n]       : 0.0f;
    w.y = (k0 + 1 < N1 && n < N2) ? W2[(k0 + 1) * N2 + n] : 0.0f;
    *(v2f*)&w2t[p * W2PP + n * 2] = w;
  }
  for (int i = tid; i < (N2P / 2) * N3P; i += 128) {      // 2 iters
    int p = i >> 4, n = i & 15;
    int k0 = 2 * p;
    v2f w;
    w.x = (k0     < N2 && n < N3) ? W3[k0 * N3 + n]       : 0.0f;
    w.y = (k0 + 1 < N2 && n < N3) ? W3[(k0 + 1) * N3 + n] : 0.0f;
    *(v2f*)&w3t[p * W3PP + n * 2] = w;
  }
  for (int i = tid; i < (N3P / 2) * NOP; i += 128) {      // 1 iter
    int p = i >> 4, n = i & 15;
    int k0 = 2 * p;
    v2f w;
    w.x = (k0     < NO && n < NO) ? Wo[k0 * NO + n]       : 0.0f;
    w.y = (k0 + 1 < NO && n < NO) ? Wo[(k0 + 1) * NO + n] : 0.0f;
    *(v2f*)&wot[p * WOPP + n * 2] = w;
  }
  if (tid < N1P) { bb1[tid] = (tid < N1) ? b1[tid] : 0.0f; aa1[tid] = (tid < N1) ? a1[tid] : 0; }
  { int n = tid - N1P;
    if (n >= 0 && n < N2P) { bb2[n] = (n < N2) ? b2[n] : 0.0f; aa2[n] = (n < N2) ? a2[n] : 0; } }
  { int n = tid - (N1P + N2P);
    if (n >= 0 && n < N3P) { bb3[n] = (n < N3) ? b3[n] : 0.0f; aa3[n] = (n < N3) ? a3[n] : 0; } }
  { int n = tid - (N1P + N2P + N3P);
    if (n >= 0 && n < NOP) { bbo[n] = (n < NO) ? bo[n] : 0.0f; } }

  // ---------------- layer 1: [64 x 784] @ [784 x 64p], 7 K-chunks -----------------
  v8f acc1[4];
  const v8f vz = {};
  acc1[0] = vz; acc1[1] = vz; acc1[2] = vz; acc1[3] = vz;

  for (int kc = 0; kc < NCH; ++kc) {
    const int kbase = kc * KC;
    __syncthreads();   // previous chunk fully consumed before restaging

    // x chunk: b128 coalesced global -> b128 LDS
    for (int i = tid; i < MB * (KC / 4); i += 128) {      // 14 iters
      int r  = i / (KC / 4);
      int c4 = (i - r * (KC / 4)) << 2;
      *(v4f*)&xs[r * XP + c4] = *(const v4f*)&x[(row0 + r) * D0 + kbase + c4];
    }
    // W1 chunk: K-pair interleaved, pad cols -> 0
    for (int i = tid; i < (KC / 2) * N1P; i += 128) {     // 28 iters
      int p = i >> 6, n = i & 63;
      long kg = kbase + 2 * p;
      v2f w;
      w.x = (n < N1) ? W1[kg * N1 + n]       : 0.0f;
      w.y = (n < N1) ? W1[(kg + 1) * N1 + n] : 0.0f;
      *(v2f*)&w1t[p * W1PP + n * 2] = w;
    }
    __syncthreads();

    // prefetch next chunk (global_prefetch_b8: no counters, pure latency hiding
    // behind the WMMA phase below)
    if (kc + 1 < NCH) {
      for (int i = tid; i < MB * 4; i += 128) {           // one per ~112B of next x chunk
        int r = i >> 2, c = (i & 3) * 28;
        __builtin_prefetch(&x[(row0 + r) * D0 + kbase + KC + c], 0, 3);
      }
      const float* wn = &W1[(long)(kbase + KC) * N1];     // next W1 chunk is contiguous
      for (int i = tid; i < (KC * N1) / 32; i += 128) {   // one per 128B
        __builtin_prefetch(wn + i * 32, 0, 3);
      }
    }

    const float* ap = &xs[(wave * 16 + n16) * XP + khi];
    const float* bp = &w1t[phi * W1PP + n16 * 2];
    #pragma unroll 4
    for (int it = 0; it < KC / 4; ++it) {
      v2f a  = *(const v2f*)(ap);                         // ds_load_b64 (2addr-fused)
      v2f b0 = *(const v2f*)(bp);                         // fragment-ready: no movs
      v2f b1 = *(const v2f*)(bp + 32);
      v2f b2 = *(const v2f*)(bp + 64);
      v2f b3 = *(const v2f*)(bp + 96);
      acc1[0] = wmma4(a, b0, acc1[0]);
      acc1[1] = wmma4(a, b1, acc1[1]);
      acc1[2] = wmma4(a, b2, acc1[2]);
      acc1[3] = wmma4(a, b3, acc1[3]);
      ap += 4;
      bp += 2 * W1PP;
    }
  }

  // epilogue layer 1: bias + per-neuron activation -> h1s (pad cols exactly 0)
  #pragma unroll
  for (int t = 0; t < 4; ++t) {
    int n = t * 16 + n16;
    float bias = bb1[n];
    int a = aa1[n];
    #pragma unroll
    for (int j = 0; j < 8; ++j) {
      int m = wave * 16 + mhi + j;
      h1s[m * H1P + n] = act_apply(a, acc1[t][j] + bias);
    }
  }
  __syncthreads();

  // ---------------- layer 2: [64 x 64p] @ [64p x 32p] -----------------------------
  v8f acc2[2]; acc2[0] = vz; acc2[1] = vz;
  {
    const float* ap = &h1s[(wave * 16 + n16) * H1P + khi];
    const float* bp = &w2t[phi * W2PP + n16 * 2];
    #pragma unroll
    for (int it = 0; it < N1P / 4; ++it) {
      v2f a  = *(const v2f*)(ap);
      v2f b0 = *(const v2f*)(bp);
      v2f b1 = *(const v2f*)(bp + 32);
      acc2[0] = wmma4(a, b0, acc2[0]);
      acc2[1] = wmma4(a, b1, acc2[1]);
      ap += 4;
      bp += 2 * W2PP;
    }
  }
  #pragma unroll
  for (int t = 0; t < 2; ++t) {
    int n = t * 16 + n16;
    float bias = bb2[n];
    int a = aa2[n];
    #pragma unroll
    for (int j = 0; j < 8; ++j) {
      int m = wave * 16 + mhi + j;
      h2s[m * H2P + n] = act_apply(a, acc2[t][j] + bias);
    }
  }
  __syncthreads();

  // ---------------- layer 3: [64 x 32p] @ [32p x 16p] -----------------------------
  v8f acc3 = vz;
  {
    const float* ap = &h2s[(wave * 16 + n16) * H2P + khi];
    const float* bp = &w3t[phi * W3PP + n16 * 2];
    #pragma unroll
    for (int it = 0; it < N2P / 4; ++it) {
      v2f a = *(const v2f*)(ap);
      v2f b = *(const v2f*)(bp);
      acc3 = wmma4(a, b, acc3);
      ap += 4;
      bp += 2 * W3PP;
    }
  }
  {
    float bias = bb3[n16];
    int a = aa3[n16];
    #pragma unroll
    for (int j = 0; j < 8; ++j) {
      int m = wave * 16 + mhi + j;
      h3s[m * H3P + n16] = act_apply(a, acc3[j] + bias);
    }
  }
  __syncthreads();

  // ---------------- output: [64 x 16p] @ [16p x 16p] + bo --------------------------
  v8f acco = vz;
  {
    const float* ap = &h3s[(wave * 16 + n16) * H3P + khi];
    const float* bp = &wot[phi * WOPP + n16 * 2];
    #pragma unroll
    for (int it = 0; it < N3P / 4; ++it) {
      v2f a = *(const v2f*)(ap);
      v2f b = *(const v2f*)(bp);
      acco = wmma4(a, b, acco);
      ap += 4;
      bp += 2 * WOPP;
    }
  }
  if (n16 < NO) {
    float bias = bbo[n16];
    #pragma unroll
    for (int j = 0; j < 8; ++j) {
      long m = row0 + wave * 16 + mhi + j;
      out[m * NO + n16] = acco[j] + bias;
    }
  }
}

extern "C" void kernel_launch(void* const* d_in, const int* in_sizes, int n_in,
                              void* d_out, int out_size, void* d_ws, size_t ws_size,
                              hipStream_t stream) {
  const float* x  = (const float*)d_in[0];
  const float* W1 = (const float*)d_in[1];
  const float* b1 = (const float*)d_in[2];
  const int*   a1 = (const int*)  d_in[3];
  const float* W2 = (const float*)d_in[4];
  const float* b2 = (const float*)d_in[5];
  const int*   a2 = (const int*)  d_in[6];
  const float* W3 = (const float*)d_in[7];
  const float* b3 = (const float*)d_in[8];
  const int*   a3 = (const int*)  d_in[9];
  const float* Wo = (const float*)d_in[10];
  const float* bo = (const float*)d_in[11];
  float* out = (float*)d_out;

  dim3 grid(B_ROWS / MB);
  dim3 block(128);
  mlp4_wmma_kernel<<<grid, block, 0, stream>>>(x, W1, b1, a1, W2, b2, a2,
                                               W3, b3, a3, Wo, bo, out);
}